// CluStream_33517924778076
// MI455X (gfx1250) — compile-verified
//
#include <hip/hip_runtime.h>
#include <math.h>

typedef __attribute__((ext_vector_type(2))) float v2f;
typedef __attribute__((ext_vector_type(8))) float v8f;

#define DDIM     512
#define KCLUST   1024
#define NROWS    16384
#define TEMP     0.07f
#define ASTRIDE  516                      // 512 + 4 pad -> LDS bank-conflict-free
#define CLOSEST_OFF (16384 * 512)
#define LOSS_OFF    (16384 * 512 + 16384)

// ---------------------------------------------------------------------------
// Kernel 1: per-cluster squared norms ||c_n||^2  (wave per cluster)
// ---------------------------------------------------------------------------
__global__ __launch_bounds__(256) void CluStream_cn2_kernel(
    const float* __restrict__ clusters, float* __restrict__ cn2) {
  const int wave = (blockIdx.x * 256 + threadIdx.x) >> 5;   // 128 blocks * 8 = 1024
  const int lane = threadIdx.x & 31;
  const float* cp = clusters + (size_t)wave * DDIM;
  float s = 0.f;
  #pragma unroll
  for (int k = lane; k < DDIM; k += 32) { float v = cp[k]; s += v * v; }
  #pragma unroll
  for (int m = 16; m > 0; m >>= 1) s += __shfl_xor(s, m, 32);
  if (lane == 0) cn2[wave] = s;
}

// ---------------------------------------------------------------------------
// Kernel 2: fused  dot = z @ clusters^T  (f32 WMMA), argmin distance,
//           z_n write-out, online log-softmax, per-block loss partial.
// Block: 256 threads (8 waves), owns 16 rows of z. Wave w owns columns
// [w*128, w*128+128) as 8 sequential 16x16 WMMA tiles over K=512.
// A tile staged into LDS with GLOBAL_LOAD_ASYNC_TO_LDS_B128 (ASYNCcnt path).
// ---------------------------------------------------------------------------
__global__ __launch_bounds__(256) void CluStream_fused_kernel(
    const float* __restrict__ z, const float* __restrict__ clusters,
    const float* __restrict__ cn2, float* __restrict__ out,
    float* __restrict__ lossPart) {
  __shared__ float As[16 * ASTRIDE];      // 16 rows of z, padded stride
  __shared__ float red[16][16];           // norm partials / loss scratch
  __shared__ float rnorm[16];
  __shared__ float part[16][8][5];        // per-row per-wave partial stats

  const int tid = threadIdx.x;
  const int m0  = blockIdx.x * 16;        // global row base

  // ---- Stage A tile (16x512 f32 = 32KB) into LDS via async-to-LDS B128 ----
  // Generic __shared__ pointer low 32 bits == wave-relative LDS byte offset.
  {
    const float* gsrc = z + (size_t)m0 * DDIM;
    #pragma unroll
    for (int i = tid; i < 16 * 128; i += 256) {
      const int row = i >> 7, c4 = i & 127;
      const unsigned lds_off =
          (unsigned)(uintptr_t)(As + row * ASTRIDE) + (unsigned)c4 * 16u;
      const float* gp = gsrc + i * 4;
      asm volatile("global_load_async_to_lds_b128 %0, %1, off"
                   :: "v"(lds_off), "v"(gp) : "memory");
    }
    asm volatile("s_wait_asynccnt 0x0" ::: "memory");
  }
  __syncthreads();

  // ---- Row norms: 16 threads per row ----
  {
    const int row = tid >> 4, seg = tid & 15;
    const float* ap = As + row * ASTRIDE + seg * 32;
    float s = 0.f;
    #pragma unroll
    for (int j = 0; j < 32; ++j) { float v = ap[j]; s += v * v; }
    red[row][seg] = s;
  }
  __syncthreads();
  if (tid < 16) {
    float t = 0.f;
    #pragma unroll
    for (int s = 0; s < 16; ++s) t += red[tid][s];
    rnorm[tid] = sqrtf(t);
  }
  __syncthreads();

  // ---- Write z_n = z / max(||z||,1e-12), coalesced float4 ----
  {
    float4* dst = (float4*)(out + (size_t)m0 * DDIM);
    #pragma unroll
    for (int i = tid; i < 16 * 128; i += 256) {
      const int row = i >> 7, c4 = i & 127;
      const float inv = 1.0f / fmaxf(rnorm[row], 1e-12f);
      float4 v = ((const float4*)(As + row * ASTRIDE))[c4];
      v.x *= inv; v.y *= inv; v.z *= inv; v.w *= inv;
      dst[i] = v;
    }
  }

  // ---- WMMA matmul + online per-row stats ----
  const int lane = tid & 31;
  const int wave = tid >> 5;              // 0..7
  const int lid  = lane & 15;             // N within tile / row within half
  const int half = lane >> 4;             // C-layout: rows r+8*half

  float sinv[8];
  #pragma unroll
  for (int r = 0; r < 8; ++r)
    sinv[r] = 1.0f / (fmaxf(rnorm[8 * half + r], 1e-12f) * TEMP);

  float bestD[8], bestX[8], lm[8], ls[8];
  int bestI[8];
  #pragma unroll
  for (int r = 0; r < 8; ++r) {
    bestD[r] = -__builtin_inff(); bestX[r] = 0.f; bestI[r] = 0x7fffffff;
    lm[r]    = -__builtin_inff(); ls[r]   = 0.f;
  }

  const float* Ap = As + lid * ASTRIDE + 2 * half;   // A frag base (LDS)
  for (int t = 0; t < 8; ++t) {
    const int n0  = wave * 128 + t * 16;
    const int col = n0 + lid;
    const float* Bp = clusters + (size_t)col * DDIM + 2 * half;

    // Warm next tile's B rows (global_prefetch_b8) while this tile computes.
    if (t < 7) {
      const float* Bn = clusters + (size_t)(col + 16) * DDIM;
      __builtin_prefetch(Bn, 0, 3);
      __builtin_prefetch(Bn + 256, 0, 3);
    }

    v8f c = {};
    #pragma unroll 8
    for (int k0 = 0; k0 < DDIM; k0 += 4) {
      v2f a = *(const v2f*)(Ap + k0);                // A: K = k0+2h, k0+2h+1
      v2f b = *(const v2f*)(Bp + k0);                // B: clusters[col][same K]
      c = __builtin_amdgcn_wmma_f32_16x16x4_f32(
          false, a, false, b, (short)0, c, false, false);
    }
    const float cw = cn2[col];
    #pragma unroll
    for (int r = 0; r < 8; ++r) {
      const float x   = c[r];
      const float adj = x - 0.5f * cw;               // argmin dist == argmax adj
      if (adj > bestD[r]) { bestD[r] = adj; bestI[r] = col; bestX[r] = x; }
      const float l  = x * sinv[r];
      const float nm = fmaxf(lm[r], l);
      ls[r] = ls[r] * __expf(lm[r] - nm) + __expf(l - nm);
      lm[r] = nm;
    }
  }

  // ---- Butterfly reduce across the 16 lanes of each half (width 16) ----
  #pragma unroll
  for (int m = 1; m < 16; m <<= 1) {
    #pragma unroll
    for (int r = 0; r < 8; ++r) {
      const float od = __shfl_xor(bestD[r], m, 16);
      const int   oi = __shfl_xor(bestI[r], m, 16);
      const float ox = __shfl_xor(bestX[r], m, 16);
      if (od > bestD[r] || (od == bestD[r] && oi < bestI[r])) {
        bestD[r] = od; bestI[r] = oi; bestX[r] = ox;
      }
      const float om = __shfl_xor(lm[r], m, 16);
      const float os = __shfl_xor(ls[r], m, 16);
      const float nm = fmaxf(lm[r], om);
      ls[r] = ls[r] * __expf(lm[r] - nm) + os * __expf(om - nm);
      lm[r] = nm;
    }
  }
  if (lid == 0) {
    #pragma unroll
    for (int r = 0; r < 8; ++r) {
      const int row = 8 * half + r;
      part[row][wave][0] = bestD[r];
      part[row][wave][1] = __int_as_float(bestI[r]);
      part[row][wave][2] = bestX[r];
      part[row][wave][3] = lm[r];
      part[row][wave][4] = ls[r];
    }
  }
  __syncthreads();

  // ---- Final per-row merge across the 8 waves (ascending column blocks) ----
  if (tid < 16) {
    const int row = tid;
    float bd = -__builtin_inff(), bx = 0.f, m = -__builtin_inff(), s = 0.f;
    int bi = 0x7fffffff;
    #pragma unroll
    for (int w = 0; w < 8; ++w) {
      const float d = part[row][w][0];
      const int   i = __float_as_int(part[row][w][1]);
      const float x = part[row][w][2];
      if (d > bd || (d == bd && i < bi)) { bd = d; bi = i; bx = x; }
      const float om = part[row][w][3], os = part[row][w][4];
      const float nm = fmaxf(m, om);
      s = s * __expf(m - nm) + os * __expf(om - nm);
      m = nm;
    }
    const float sinvr = 1.0f / (fmaxf(rnorm[row], 1e-12f) * TEMP);
    const float logZ  = m + __logf(s);
    out[CLOSEST_OFF + m0 + row] = (float)bi;
    red[0][row] = -(bx * sinvr - logZ);              // per-row NLL
  }
  __syncthreads();
  if (tid == 0) {
    float t = 0.f;
    #pragma unroll
    for (int r = 0; r < 16; ++r) t += red[0][r];
    lossPart[blockIdx.x] = t;
  }
}

// ---------------------------------------------------------------------------
// Kernel 3: deterministic fixed-order reduction of 1024 loss partials
// ---------------------------------------------------------------------------
__global__ __launch_bounds__(256) void CluStream_loss_kernel(
    const float* __restrict__ lossPart, float* __restrict__ out) {
  __shared__ float sh[256];
  const int t = threadIdx.x;
  sh[t] = lossPart[t] + lossPart[t + 256] + lossPart[t + 512] + lossPart[t + 768];
  __syncthreads();
  for (int w = 128; w > 0; w >>= 1) {
    if (t < w) sh[t] += sh[t + w];
    __syncthreads();
  }
  if (t == 0) out[LOSS_OFF] = sh[0] / (float)NROWS;
}

// ---------------------------------------------------------------------------
extern "C" void kernel_launch(void* const* d_in, const int* in_sizes, int n_in,
                              void* d_out, int out_size, void* d_ws, size_t ws_size,
                              hipStream_t stream) {
  const float* z        = (const float*)d_in[0];   // [16384, 512] f32
  const float* clusters = (const float*)d_in[1];   // [1024, 512] f32
  float* out = (float*)d_out;                      // z_n | closest(as f32) | loss
  float* ws  = (float*)d_ws;
  float* lossPart = ws;                            // 1024 floats
  float* cn2      = ws + 1024;                     // 1024 floats

  CluStream_cn2_kernel<<<128, 256, 0, stream>>>(clusters, cn2);
  CluStream_fused_kernel<<<NROWS / 16, 256, 0, stream>>>(z, clusters, cn2, out, lossPart);
  CluStream_loss_kernel<<<1, 256, 0, stream>>>(lossPart, out);
}